// GCNGraphLev_62130996904046
// MI455X (gfx1250) — compile-verified
//
#include <hip/hip_runtime.h>
#include <math.h>
#include <stdint.h>

#define DHID 128
#define DOUTC 10
#define NGRAPH 128

typedef __attribute__((ext_vector_type(2))) float v2f;
typedef __attribute__((ext_vector_type(8))) float v8f;
typedef __attribute__((ext_vector_type(4))) unsigned int u32x4;
typedef __attribute__((ext_vector_type(8))) int i32x8;
typedef __attribute__((ext_vector_type(4))) int i32x4;

// ---------------- degree / norm ----------------
__global__ __launch_bounds__(256) void k_fill1(float* __restrict__ p, int n) {
  int i = blockIdx.x * 256 + threadIdx.x;
  if (i < n) p[i] = 1.0f;  // self-loop contributes 1 to degree
}

__global__ __launch_bounds__(256) void k_deg(const int* __restrict__ dst,
                                             float* __restrict__ deg, int e) {
  int i = blockIdx.x * 256 + threadIdx.x;
  if (i < e) atomicAdd(&deg[__builtin_nontemporal_load(&dst[i])], 1.0f);
}

__global__ __launch_bounds__(256) void k_rsqrt(float* __restrict__ p, int n) {
  int i = blockIdx.x * 256 + threadIdx.x;
  if (i < n) p[i] = rsqrtf(p[i]);
}

// ---------------- fp32 WMMA GEMM: H[N,128] = (relu?)X[N,128] @ W[128,128] ----------------
// One block: 16 rows x all 128 cols. TDM stages the 16x128 A tile into LDS once;
// each of the 8 waves computes one 16x16 tile with V_WMMA_F32_16X16X4_F32.
template <bool RELU>
__global__ __launch_bounds__(256) void k_gemm(const float* __restrict__ X,
                                              const float* __restrict__ W,
                                              float* __restrict__ H) {
  __shared__ float tile[16 * DHID];  // 8 KB A tile

  const int lane = threadIdx.x & 31;
  const int wave = threadIdx.x >> 5;
  const int m0 = blockIdx.x * 16;   // row block
  const int n0 = wave * 16;         // col block
  const int r15 = lane & 15;        // A row / B,D col within tile
  const int kh = lane >> 4;         // K-half selector

  if (wave == 0) {
    // Tensor DMA descriptor (D#): 2-D tensor, 4-byte elements,
    // tile = 128 (dim0, contiguous) x 16 (dim1 rows), stride0 = 128 elements.
    unsigned long long ga =
        (unsigned long long)(uintptr_t)(X + (size_t)m0 * DHID);
    unsigned lds_off = (unsigned)(uintptr_t)&tile[0];  // addr[31:0] == LDS offset
    u32x4 g0;
    g0[0] = 1u;                                   // count=1, user descriptor
    g0[1] = lds_off;                              // lds_addr
    g0[2] = (unsigned)(ga & 0xFFFFFFFFu);         // global_addr[31:0]
    g0[3] = (unsigned)((ga >> 32) & 0x01FFFFFFu)  // global_addr[56:32]
            | (2u << 30);                         // type = 2 ("image")
    i32x8 g1;
    g1[0] = 0x00020000;           // workgroup_mask=0, data_size=2 (4B)
    g1[1] = (int)(128u << 16);    // tensor_dim0[15:0] = 128
    g1[2] = (int)(16u << 16);     // tensor_dim0 hi=0 | tensor_dim1[15:0]=16
    g1[3] = (int)(128u << 16);    // tensor_dim1 hi=0 | tile_dim0=128
    g1[4] = 16;                   // tile_dim1=16, tile_dim2=0 (unused)
    g1[5] = 128;                  // tensor_dim0_stride lo = 128
    g1[6] = 0;                    // stride0 hi | stride1 lo
    g1[7] = 0;                    // stride1 hi
    i32x4 gz = {0, 0, 0, 0};      // groups 2/3 unused (2-D tensor)
    i32x8 gz8 = {0, 0, 0, 0, 0, 0, 0, 0};
    __builtin_amdgcn_tensor_load_to_lds(g0, g1, gz, gz, gz8, 0);
    __builtin_amdgcn_s_wait_tensorcnt(0);
  }
  __syncthreads();

  const float* __restrict__ arow = &tile[r15 * DHID];
  v8f c = {};
  for (int k0 = 0; k0 < DHID; k0 += 4) {
    const int ka = k0 + 2 * kh;
    v2f a = *(const v2f*)&arow[ka];  // ds_load_b64
    if (RELU) {
      a.x = fmaxf(a.x, 0.0f);
      a.y = fmaxf(a.y, 0.0f);
    }
    v2f b;
    b.x = W[(size_t)ka * DHID + n0 + r15];
    b.y = W[(size_t)(ka + 1) * DHID + n0 + r15];
    c = __builtin_amdgcn_wmma_f32_16x16x4_f32(false, a, false, b,
                                              (short)0, c, false, false);
  }
  float* __restrict__ hb = H + (size_t)m0 * DHID + n0 + r15;
#pragma unroll
  for (int r = 0; r < 8; ++r) {
    hb[(size_t)(r + 8 * kh) * DHID] = c[r];
  }
}

// ---------------- self-loop + bias init: OUT[i,:] = dinv[i]^2 * H[i,:] + b ----------------
__global__ __launch_bounds__(256) void k_init(const float* __restrict__ H,
                                              const float* __restrict__ dinv,
                                              const float* __restrict__ bias,
                                              float* __restrict__ OUT, int n) {
  int idx = blockIdx.x * 256 + threadIdx.x;  // over n * 32 float4s
  if (idx >= n * (DHID / 4)) return;
  int i = idx >> 5;
  int j4 = idx & 31;
  float di = dinv[i];
  float d2 = di * di;
  float4 h = ((const float4*)H)[idx];
  float4 b = ((const float4*)bias)[j4];
  float4 o;
  o.x = d2 * h.x + b.x;
  o.y = d2 * h.y + b.y;
  o.z = d2 * h.z + b.z;
  o.w = d2 * h.w + b.w;
  ((float4*)OUT)[idx] = o;
}

// ---------------- edge scatter: OUT[d,:] += dinv[s]*dinv[d]*H[s,:] (1 wave / edge) --------
__global__ __launch_bounds__(256) void k_scatter(const int* __restrict__ src,
                                                 const int* __restrict__ dst,
                                                 const float* __restrict__ dinv,
                                                 const float* __restrict__ H,
                                                 float* __restrict__ OUT, int e) {
  int eid = blockIdx.x * 8 + (threadIdx.x >> 5);
  if (eid >= e) return;
  int lane = threadIdx.x & 31;
  int s = __builtin_nontemporal_load(&src[eid]);  // edge lists streamed once -> NT
  int d = __builtin_nontemporal_load(&dst[eid]);
  float w = dinv[s] * dinv[d];
  float4 h = ((const float4*)(H + (size_t)s * DHID))[lane];  // global_load_b128
  float* od = OUT + (size_t)d * DHID + lane * 4;
  atomicAdd(od + 0, w * h.x);
  atomicAdd(od + 1, w * h.y);
  atomicAdd(od + 2, w * h.z);
  atomicAdd(od + 3, w * h.w);
}

// ---------------- pooling ----------------
__global__ __launch_bounds__(256) void k_zero(float* __restrict__ p, int n) {
  int i = blockIdx.x * 256 + threadIdx.x;
  if (i < n) p[i] = 0.0f;
}

__global__ __launch_bounds__(256) void k_pool(const int* __restrict__ batch,
                                              const float* __restrict__ act,
                                              float* __restrict__ pooled,
                                              float* __restrict__ cnt, int n) {
  int idx = blockIdx.x * 256 + threadIdx.x;  // over n * 32 float4s
  if (idx >= n * (DHID / 4)) return;
  int i = idx >> 5;
  int j4 = idx & 31;
  int g = batch[i];
  float4 a = ((const float4*)act)[idx];
  float* p = &pooled[(size_t)g * DHID + j4 * 4];
  atomicAdd(p + 0, a.x);
  atomicAdd(p + 1, a.y);
  atomicAdd(p + 2, a.z);
  atomicAdd(p + 3, a.w);
  if (j4 == 0) atomicAdd(&cnt[g], 1.0f);
}

// ---------------- head: logits = (pooled/cnt) @ Wl + bl; log_softmax ----------------
__global__ __launch_bounds__(NGRAPH) void k_head(const float* __restrict__ pooled,
                                                 const float* __restrict__ cnt,
                                                 const float* __restrict__ Wl,
                                                 const float* __restrict__ bl,
                                                 float* __restrict__ out) {
  int g = threadIdx.x;  // one thread per graph, G == 128
  float inv = 1.0f / fmaxf(cnt[g], 1.0f);
  float logits[DOUTC];
#pragma unroll
  for (int d = 0; d < DOUTC; ++d) logits[d] = bl[d];
  for (int k = 0; k < DHID; ++k) {
    float p = pooled[(size_t)g * DHID + k] * inv;
#pragma unroll
    for (int d = 0; d < DOUTC; ++d) logits[d] += p * Wl[k * DOUTC + d];
  }
  float m = logits[0];
#pragma unroll
  for (int d = 1; d < DOUTC; ++d) m = fmaxf(m, logits[d]);
  float s = 0.0f;
#pragma unroll
  for (int d = 0; d < DOUTC; ++d) s += __expf(logits[d] - m);
  float lse = m + __logf(s);
#pragma unroll
  for (int d = 0; d < DOUTC; ++d) out[(size_t)g * DOUTC + d] = logits[d] - lse;
}

extern "C" void kernel_launch(void* const* d_in, const int* in_sizes, int n_in,
                              void* d_out, int out_size, void* d_ws, size_t ws_size,
                              hipStream_t stream) {
  const float* x   = (const float*)d_in[0];
  const int* ei    = (const int*)d_in[1];
  const int* batch = (const int*)d_in[2];
  const float* W1  = (const float*)d_in[3];
  const float* b1  = (const float*)d_in[4];
  const float* W2  = (const float*)d_in[5];
  const float* b2  = (const float*)d_in[6];
  const float* W3  = (const float*)d_in[7];
  const float* b3  = (const float*)d_in[8];
  const float* Wl  = (const float*)d_in[9];
  const float* bl  = (const float*)d_in[10];

  const int N_ = in_sizes[0] / DHID;  // 50000 (multiple of 16)
  const int E_ = in_sizes[1] / 2;     // 800000
  const int* src = ei;                // edge_index row 0
  const int* dst = ei + E_;           // edge_index row 1

  // workspace layout (floats): dinv[N] | H[N*128] | A[N*128] | pooled[G*128] | cnt[G]
  float* ws = (float*)d_ws;
  float* dinv = ws;
  size_t off = (size_t)((N_ + 255) & ~255);
  float* H = ws + off;
  float* A = H + (size_t)N_ * DHID;
  float* pooled = A + (size_t)N_ * DHID;
  float* cnt = pooled + (size_t)NGRAPH * DHID;  // contiguous after pooled

  const int gN  = (N_ + 255) / 256;
  const int gE  = (E_ + 255) / 256;
  const int gMM = N_ / 16;                       // WMMA grid: 16 rows per block
  const int gV4 = (N_ * (DHID / 4) + 255) / 256; // float4 elementwise grid
  const int gSC = (E_ + 7) / 8;                  // 8 edges (waves) per block

  // symmetric normalization: dinv = rsqrt(1 + indegree)
  k_fill1<<<gN, 256, 0, stream>>>(dinv, N_);
  k_deg<<<gE, 256, 0, stream>>>(dst, dinv, E_);
  k_rsqrt<<<gN, 256, 0, stream>>>(dinv, N_);

  // layer 1
  k_gemm<false><<<gMM, 256, 0, stream>>>(x, W1, H);
  k_init<<<gV4, 256, 0, stream>>>(H, dinv, b1, A, N_);
  k_scatter<<<gSC, 256, 0, stream>>>(src, dst, dinv, H, A, E_);
  // layer 2 (relu fused into GEMM A-load)
  k_gemm<true><<<gMM, 256, 0, stream>>>(A, W2, H);
  k_init<<<gV4, 256, 0, stream>>>(H, dinv, b2, A, N_);
  k_scatter<<<gSC, 256, 0, stream>>>(src, dst, dinv, H, A, E_);
  // layer 3
  k_gemm<true><<<gMM, 256, 0, stream>>>(A, W3, H);
  k_init<<<gV4, 256, 0, stream>>>(H, dinv, b3, A, N_);
  k_scatter<<<gSC, 256, 0, stream>>>(src, dst, dinv, H, A, E_);

  // global mean pool + classifier head
  k_zero<<<(NGRAPH * DHID + NGRAPH + 255) / 256, 256, 0, stream>>>(
      pooled, NGRAPH * DHID + NGRAPH);
  k_pool<<<gV4, 256, 0, stream>>>(batch, A, pooled, cnt, N_);
  k_head<<<1, NGRAPH, 0, stream>>>(pooled, cnt, Wl, bl, (float*)d_out);
}